// myGCN_4157528342727
// MI455X (gfx1250) — compile-verified
//
#include <hip/hip_runtime.h>
#include <hip/hip_bf16.h>

typedef __attribute__((ext_vector_type(2))) float v2f;
typedef __attribute__((ext_vector_type(8))) float v8f;

#define HID 64  // both layer output dims are 64

// ---------------------------------------------------------------------------
// 1) init: deg[i] = 1.0 (self-loop weight), seed min/max
// ---------------------------------------------------------------------------
__global__ void init_kernel(float* __restrict__ deg, int* __restrict__ mm, int N) {
    int i = blockIdx.x * blockDim.x + threadIdx.x;
    if (i < N) deg[i] = 1.0f;
    if (i == 0) { mm[0] = 0x7FFFFFFF; mm[1] = 0x80000000; }
}

// ---------------------------------------------------------------------------
// 2) min/max of edge_count (grid-stride + one atomic pair per thread)
// ---------------------------------------------------------------------------
__global__ void minmax_kernel(const int* __restrict__ ec, int E, int* __restrict__ mm) {
    int i = blockIdx.x * blockDim.x + threadIdx.x;
    int stride = gridDim.x * blockDim.x;
    int lmin = 0x7FFFFFFF, lmax = 0x80000000;
    for (; i < E; i += stride) {
        int v = ec[i];
        lmin = min(lmin, v);
        lmax = max(lmax, v);
    }
    atomicMin(&mm[0], lmin);
    atomicMax(&mm[1], lmax);
}

// ---------------------------------------------------------------------------
// 3) edge weight normalization + degree accumulation (by col)
// ---------------------------------------------------------------------------
__global__ void edgew_kernel(const int* __restrict__ ec, const int* __restrict__ col,
                             const int* __restrict__ mm, float* __restrict__ ew,
                             float* __restrict__ deg, int E) {
    int e = blockIdx.x * blockDim.x + threadIdx.x;
    if (e >= E) return;
    float mn = (float)mm[0], mx = (float)mm[1];
    float denom = mx - mn;
    float w = (denom > 0.0f) ? ((float)ec[e] - mn) / denom : 0.0f;
    ew[e] = w;
    atomicAdd(&deg[col[e]], w);
}

// ---------------------------------------------------------------------------
// 4) dinv = rsqrt(deg) in place  (deg >= 1 always, guard anyway)
// ---------------------------------------------------------------------------
__global__ void dinv_kernel(float* __restrict__ deg, int N) {
    int i = blockIdx.x * blockDim.x + threadIdx.x;
    if (i >= N) return;
    float d = deg[i];
    deg[i] = (d > 0.0f) ? rsqrtf(d) : 0.0f;
}

// ---------------------------------------------------------------------------
// 5/8) fp32 WMMA GEMM:  H[M,64] = act(A[M,K]) @ W[K,64]
//      One wave -> one 16x64 output tile via 4 accumulators of
//      v_wmma_f32_16x16x4_f32.
//
//      W is staged in LDS in K-PAIR-SWIZZLED layout:
//        sW[(k>>1)*128 + 2*c + (k&1)]  ==  W[k][c]
//      so a lane's B operand {W[kb][c], W[kb+1][c]} is one contiguous,
//      8-byte-aligned ds_load_b64 (no register shuffles, no 2addr loads).
//
//      A-operand layout (ISA 7.12.2, 32-bit A 16x4):
//        lanes 0-15: row M=lane,    VGPR0=K+0, VGPR1=K+1
//        lanes16-31: row M=lane-16, VGPR0=K+2, VGPR1=K+3
//      B-operand: row K striped across lanes (N=lane&15), halves hold K,K+2.
// ---------------------------------------------------------------------------
template <int K, bool RELU>
__global__ void gemm_wmma_kernel(const float* __restrict__ A, const float* __restrict__ W,
                                 float* __restrict__ H, int M) {
    __shared__ __align__(16) float sW[K * HID];

    // cooperative swizzled load: pair rows (2k, 2k+1) interleaved per column
    for (int i = threadIdx.x; i < (K / 2) * HID; i += blockDim.x) {
        int kp = i >> 6;          // k-pair index
        int c  = i & 63;          // column
        float w0 = W[(2 * kp) * HID + c];
        float w1 = W[(2 * kp + 1) * HID + c];
        float* dst = sW + kp * (2 * HID) + 2 * c;
        dst[0] = w0;
        dst[1] = w1;
    }
    __syncthreads();

    const int lane  = threadIdx.x & 31;
    const int widx  = threadIdx.x >> 5;             // wave in block (0..7)
    const int half  = lane >> 4;                    // 0: lanes 0-15, 1: lanes 16-31
    const int sub   = lane & 15;                    // M row (A) / N col (B,C,D)
    const long tile = (long)blockIdx.x * 8 + widx;  // 16-row tile index
    const long row0 = tile * 16;
    if (row0 >= M) return;                          // no syncs after this point

    v8f acc[4] = {};                                // 4 x (16x16 f32) = 16x64

    const float* aRow = A + (row0 + sub) * (long)K;

#pragma unroll 4
    for (int kk = 0; kk < K; kk += 4) {
        const int kb = kk + 2 * half;               // this half's first K index (even)
        // A operand: two consecutive K values for this lane's row (b64 load)
        v2f a = *(const v2f*)(aRow + kb);
        if (RELU) { a.x = fmaxf(a.x, 0.0f); a.y = fmaxf(a.y, 0.0f); }

        // B operand base: pair (kb, kb+1) at column sub (+ t*16)
        const float* wp = sW + (kb >> 1) * (2 * HID) + 2 * sub;
#pragma unroll
        for (int t = 0; t < 4; ++t) {               // N tiles 0,16,32,48
            v2f b = *(const v2f*)(wp + t * 32);     // single ds_load_b64
            acc[t] = __builtin_amdgcn_wmma_f32_16x16x4_f32(
                /*neg_a=*/false, a, /*neg_b=*/false, b,
                /*c_mod=*/(short)0, acc[t],
                /*reuse_a=*/false, /*reuse_b=*/false);
        }
    }

    // C/D layout: VGPR r -> row r (lanes 0-15) / row r+8 (lanes 16-31), col = sub
#pragma unroll
    for (int r = 0; r < 8; ++r) {
        float* hp = H + (row0 + r + 8 * half) * (long)HID + sub;
#pragma unroll
        for (int t = 0; t < 4; ++t) hp[t * 16] = acc[t][r];
    }
}

// ---------------------------------------------------------------------------
// 6/9) self-loop term + bias (full overwrite -> replay-safe init of aggregate)
//      out[i,d] = h[i,d] * dinv[i]^2 + bias[d]
// ---------------------------------------------------------------------------
__global__ void selfloop_bias_kernel(const float* __restrict__ h, const float* __restrict__ dinv,
                                     const float* __restrict__ bias, float* __restrict__ out,
                                     long total) {
    long t = (long)blockIdx.x * blockDim.x + threadIdx.x;
    if (t >= total) return;
    long i = t >> 6;
    int  d = (int)(t & 63);
    float di = dinv[i];
    out[t] = h[t] * di * di + bias[d];
}

// ---------------------------------------------------------------------------
// 7/10) edge scatter: one wave per edge; lanes cover 64 dims (lane, lane+32).
//       out[col] += h[row] * (dinv[row]*ew*dinv[col])   via f32 atomics (L2)
// ---------------------------------------------------------------------------
__global__ void scatter_kernel(const float* __restrict__ h, const float* __restrict__ dinv,
                               const float* __restrict__ ew, const int* __restrict__ row,
                               const int* __restrict__ col, float* __restrict__ out, int E) {
    const int lane = threadIdx.x & 31;
    const int widx = threadIdx.x >> 5;       // 8 waves per block
    const long ebase = (long)blockIdx.x * 32;
#pragma unroll
    for (int k = 0; k < 4; ++k) {
        long e = ebase + (long)k * 8 + widx;
        if (e >= E) continue;
        int r = row[e], c = col[e];
        float norm = dinv[r] * ew[e] * dinv[c];
        const float* hr = h + (long)r * HID;
        float v0 = hr[lane]      * norm;
        float v1 = hr[lane + 32] * norm;
        float* op = out + (long)c * HID;
        atomicAdd(op + lane,      v0);
        atomicAdd(op + lane + 32, v1);
    }
}

// ---------------------------------------------------------------------------
// 11) log_softmax over dim=64: one wave per node, shfl_xor reductions
// ---------------------------------------------------------------------------
__global__ void logsoftmax_kernel(const float* __restrict__ H, float* __restrict__ out, int N) {
    int gid  = blockIdx.x * blockDim.x + threadIdx.x;
    int node = gid >> 5;
    int lane = gid & 31;
    if (node >= N) return;
    const float* rowp = H + (long)node * HID;
    float a = rowp[lane];
    float b = rowp[lane + 32];
    float m = fmaxf(a, b);
#pragma unroll
    for (int off = 16; off > 0; off >>= 1) m = fmaxf(m, __shfl_xor(m, off, 32));
    float s = __expf(a - m) + __expf(b - m);
#pragma unroll
    for (int off = 16; off > 0; off >>= 1) s += __shfl_xor(s, off, 32);
    float lse = m + __logf(s);
    float* op = out + (long)node * HID;
    op[lane]      = a - lse;
    op[lane + 32] = b - lse;
}

// ---------------------------------------------------------------------------
// launcher
// ---------------------------------------------------------------------------
extern "C" void kernel_launch(void* const* d_in, const int* in_sizes, int n_in,
                              void* d_out, int out_size, void* d_ws, size_t ws_size,
                              hipStream_t stream) {
    const float* x  = (const float*)d_in[0];
    const int*   ei = (const int*)d_in[1];
    const int*   ec = (const int*)d_in[2];
    const float* W1 = (const float*)d_in[3];
    const float* b1 = (const float*)d_in[4];
    const float* W2 = (const float*)d_in[5];
    const float* b2 = (const float*)d_in[6];

    const int IN_DIM = 256;
    const int N = in_sizes[0] / IN_DIM;   // 100000
    const int E = in_sizes[2];            // 1600000
    const int* row = ei;                  // edge_index[0]
    const int* col = ei + E;              // edge_index[1]

    // workspace layout (floats)
    float* wsf  = (float*)d_ws;
    float* ew   = wsf;                          // E
    float* deg  = ew   + (size_t)E;             // N (becomes dinv in place)
    float* h    = deg  + (size_t)N;             // N*64  (GEMM1 output)
    float* agg1 = h    + (size_t)N * HID;       // N*64  (layer-1 aggregate)
    float* h2   = agg1 + (size_t)N * HID;       // N*64  (GEMM2 output)
    int*   mm   = (int*)(h2 + (size_t)N * HID); // 2 ints (min/max)

    float* emb = (float*)d_out;                 // node_embeddings [N,64]
    float* lsm = emb + (size_t)N * HID;         // log_softmax     [N,64]

    const long NH = (long)N * HID;
    const int tileBlocks = (int)((N / 16 + 7) / 8);       // 8 waves/block, 16 rows/wave
    const int edgeBlocks = (E + 31) / 32;                 // 32 edges/block

    // --- normalization pipeline ---
    init_kernel<<<(N + 255) / 256, 256, 0, stream>>>(deg, mm, N);
    minmax_kernel<<<1024, 256, 0, stream>>>(ec, E, mm);
    edgew_kernel<<<(E + 255) / 256, 256, 0, stream>>>(ec, col, mm, ew, deg, E);
    dinv_kernel<<<(N + 255) / 256, 256, 0, stream>>>(deg, N);

    // --- layer 1: h = x @ W1 ; aggregate ; (ReLU fused into GEMM2 loads) ---
    gemm_wmma_kernel<256, false><<<tileBlocks, 256, 0, stream>>>(x, W1, h, N);
    selfloop_bias_kernel<<<(int)((NH + 255) / 256), 256, 0, stream>>>(h, deg, b1, agg1, NH);
    scatter_kernel<<<edgeBlocks, 256, 0, stream>>>(h, deg, ew, row, col, agg1, E);

    // --- layer 2: h2 = relu(agg1) @ W2 ; aggregate into d_out ---
    gemm_wmma_kernel<64, true><<<tileBlocks, 256, 0, stream>>>(agg1, W2, h2, N);
    selfloop_bias_kernel<<<(int)((NH + 255) / 256), 256, 0, stream>>>(h2, deg, b2, emb, NH);
    scatter_kernel<<<edgeBlocks, 256, 0, stream>>>(h2, deg, ew, row, col, emb, E);

    // --- log_softmax into second half of d_out ---
    logsoftmax_kernel<<<(N + 7) / 8, 256, 0, stream>>>(emb, lsm, N);
}